// Net_49658411876703
// MI455X (gfx1250) — compile-verified
//
#include <hip/hip_runtime.h>
#include <hip/hip_bf16.h>

typedef __attribute__((ext_vector_type(2))) float v2f;
typedef __attribute__((ext_vector_type(8))) float v8f;

#define HDIM 32

// ---------------- utility kernels ----------------

__global__ void zero_f32(float* __restrict__ p, size_t n) {
    size_t i = (size_t)blockIdx.x * blockDim.x + threadIdx.x;
    if (i < n) p[i] = 0.0f;
}

__global__ void fill_f32(float* __restrict__ p, float v, size_t n) {
    size_t i = (size_t)blockIdx.x * blockDim.x + threadIdx.x;
    if (i < n) p[i] = v;
}

// deg[col[e]] += 1  (self-loop handled by deg init = 1.0)
__global__ void degree_kernel(const int* __restrict__ cols, float* __restrict__ deg, int E) {
    int e = blockIdx.x * blockDim.x + threadIdx.x;
    if (e < E) unsafeAtomicAdd(&deg[cols[e]], 1.0f);
}

__global__ void rsqrt_kernel(float* __restrict__ d, int n) {
    int i = blockIdx.x * blockDim.x + threadIdx.x;
    if (i < n) d[i] = rsqrtf(d[i]);  // deg >= 1 always
}

// ---------------- fp32 WMMA GEMM:  out[N,32] = act(in)[N,K] @ W[K,32] ----------------
// One wave per 16x16 output tile; blockDim=64 (2 waves cover H=32 columns).
// Uses V_WMMA_F32_16X16X4_F32 (full fp32 precision).
// FUSE_IN_BN: apply y = relu(A[k]*x + B[k]) to the A-matrix input on load
// (folds the previous layer's BatchNorm+ReLU into this GEMM, saving a full
// read+write pass over the feature matrix). ab: A at ab[0..K), B at ab[K..2K).
template <int K, bool FUSE_IN_BN>
__global__ __launch_bounds__(64) void gemm_wmma_f32(const float* __restrict__ in,
                                                    const float* __restrict__ W,
                                                    const float* __restrict__ ab,
                                                    float* __restrict__ out, int N) {
    const int wave = threadIdx.x >> 5;       // 0 or 1 -> column tile
    const int lane = threadIdx.x & 31;
    const int half = lane >> 4;              // 0: lanes 0-15, 1: lanes 16-31
    const int l    = lane & 15;
    const int row0 = blockIdx.x * 16;
    const int col0 = wave * 16;

    const float* arow = in + (size_t)(row0 + l) * K;

    v8f acc = {};
#pragma unroll
    for (int k = 0; k < K; k += 4) {
        const int k0 = k + 2 * half;
        // A-matrix 16x4 fragment (ISA 7.12.2): lanes 0-15 hold K=k,k+1 ;
        // lanes 16-31 hold K=k+2,k+3 for the same M rows.
        float a0 = arow[k0 + 0];
        float a1 = arow[k0 + 1];
        if (FUSE_IN_BN) {
            a0 = fmaxf(fmaf(ab[k0 + 0], a0, ab[K + k0 + 0]), 0.0f);
            a1 = fmaxf(fmaf(ab[k0 + 1], a1, ab[K + k0 + 1]), 0.0f);
        }
        v2f a; a.x = a0; a.y = a1;
        // B-matrix 4x16 fragment: VGPR0 = rows k (lanes 0-15) / k+2 (lanes 16-31),
        // VGPR1 = rows k+1 / k+3 ; N = col0 + l striped across lanes.
        v2f b;
        b.x = W[(k0 + 0) * HDIM + col0 + l];
        b.y = W[(k0 + 1) * HDIM + col0 + l];
        acc = __builtin_amdgcn_wmma_f32_16x16x4_f32(
            /*neg_a=*/false, a, /*neg_b=*/false, b,
            /*c_mod=*/(short)0, acc, /*reuse_a=*/false, /*reuse_b=*/false);
    }

    // C/D layout: VGPR r -> M = r (lanes 0-15), M = r+8 (lanes 16-31); N = l.
#pragma unroll
    for (int r = 0; r < 8; ++r) {
        out[(size_t)(row0 + r + 8 * half) * HDIM + col0 + l] = acc[r];
    }
}

// ---------------- edge scatter:  agg[col] += dinv[row]*dinv[col] * h[row] ----------------
// 8 threads per edge; each thread owns a float4 feature quad -> the 8 lanes of an
// edge together fetch the full 128B row (coalesced) and issue 4 fp32 atomics each.
__global__ void scatter_kernel(const int* __restrict__ rows, const int* __restrict__ cols,
                               const float* __restrict__ dinv, const float* __restrict__ h,
                               float* __restrict__ agg, int E) {
    long long tid = (long long)blockIdx.x * blockDim.x + threadIdx.x;
    int e = (int)(tid >> 3);
    if (e >= E) return;
    int q = ((int)tid & 7) << 2;  // feature quad start: 0,4,...,28
    int r = rows[e];
    int c = cols[e];
    float w = dinv[r] * dinv[c];
    const float4 v = *(const float4*)(h + (size_t)r * HDIM + q);
    float* dst = agg + (size_t)c * HDIM + q;
    unsafeAtomicAdd(dst + 0, w * v.x);
    unsafeAtomicAdd(dst + 1, w * v.y);
    unsafeAtomicAdd(dst + 2, w * v.z);
    unsafeAtomicAdd(dst + 3, w * v.w);
}

// ---------------- fused self-loop + bias + column stats ----------------
// val = agg[i,f] + dinv[i]^2 * h[i,f] + bias[f]; agg[i,f] = val;
// stats[f] += val; stats[32+f] += val*val  (block-level LDS partials first)
__global__ __launch_bounds__(256) void fuse_stats_kernel(float* __restrict__ agg,
                                                         const float* __restrict__ h,
                                                         const float* __restrict__ dinv,
                                                         const float* __restrict__ bias,
                                                         float* __restrict__ stats, int N) {
    __shared__ float s_sum[HDIM];
    __shared__ float s_sq[HDIM];
    int t = threadIdx.x;
    if (t < HDIM) { s_sum[t] = 0.0f; s_sq[t] = 0.0f; }
    __syncthreads();

    size_t idx = (size_t)blockIdx.x * blockDim.x + t;
    int f = (int)(idx & (HDIM - 1));
    bool ok = idx < (size_t)N * HDIM;
    if (ok) {
        int i = (int)(idx >> 5);
        float d = dinv[i];
        float val = agg[idx] + d * d * h[idx] + bias[f];
        agg[idx] = val;
        atomicAdd(&s_sum[f], val);        // lane == f within a wave: conflict-free
        atomicAdd(&s_sq[f], val * val);
    }
    __syncthreads();
    if (t < HDIM) {
        unsafeAtomicAdd(&stats[t], s_sum[t]);
        unsafeAtomicAdd(&stats[HDIM + t], s_sq[t]);
    }
}

// A = gamma * rsqrt(var + eps);  B = beta - mean * A  (stored at ab[0..63])
__global__ void bn_finalize_kernel(const float* __restrict__ stats, float* __restrict__ ab,
                                   const float* __restrict__ gamma, const float* __restrict__ beta,
                                   float invN) {
    int f = threadIdx.x;
    if (f >= HDIM) return;
    float mean = stats[f] * invN;
    float var  = stats[HDIM + f] * invN - mean * mean;
    float A = gamma[f] * rsqrtf(var + 1e-5f);
    ab[f] = A;
    ab[HDIM + f] = beta[f] - mean * A;
}

// Final BN apply (no relu), float4-vectorized: x[i,f] = A[f]*x[i,f] + B[f]
__global__ void bn_apply4_kernel(float* __restrict__ x, const float* __restrict__ ab, size_t nq) {
    size_t q = (size_t)blockIdx.x * blockDim.x + threadIdx.x;
    if (q >= nq) return;
    int f = (int)((q << 2) & (HDIM - 1));
    float4 v = ((float4*)x)[q];
    const float4 A = *(const float4*)(ab + f);
    const float4 B = *(const float4*)(ab + HDIM + f);
    v.x = fmaf(A.x, v.x, B.x);
    v.y = fmaf(A.y, v.y, B.y);
    v.z = fmaf(A.z, v.z, B.z);
    v.w = fmaf(A.w, v.w, B.w);
    ((float4*)x)[q] = v;
}

// ---------------- launcher ----------------

static inline int cdiv(long long a, int b) { return (int)((a + b - 1) / b); }

extern "C" void kernel_launch(void* const* d_in, const int* in_sizes, int n_in,
                              void* d_out, int out_size, void* d_ws, size_t ws_size,
                              hipStream_t stream) {
    const float* x      = (const float*)d_in[0];
    const int*   edges  = (const int*)d_in[1];
    const float* W1     = (const float*)d_in[2];
    const float* b1     = (const float*)d_in[3];
    const float* gamma1 = (const float*)d_in[4];
    const float* beta1  = (const float*)d_in[5];
    const float* W2     = (const float*)d_in[6];
    const float* b2     = (const float*)d_in[7];
    const float* gamma2 = (const float*)d_in[8];
    const float* beta2  = (const float*)d_in[9];
    float* out = (float*)d_out;

    const int N = in_sizes[0] / 128;   // F = 128
    const int E = in_sizes[1] / 2;
    const int* rows = edges;           // edge_index[0] = source
    const int* cols = edges + E;       // edge_index[1] = target

    // workspace layout (floats): dinv[N] | h[N*32] | agg[N*32] | stats[128]
    float* dinv  = (float*)d_ws;
    float* h     = dinv + N;
    float* agg   = h + (size_t)N * HDIM;
    float* stats = agg + (size_t)N * HDIM;
    const size_t NH = (size_t)N * HDIM;

    // --- symmetric normalization coefficients ---
    fill_f32<<<cdiv(N, 256), 256, 0, stream>>>(dinv, 1.0f, N);           // self loops
    degree_kernel<<<cdiv(E, 256), 256, 0, stream>>>(cols, dinv, E);
    rsqrt_kernel<<<cdiv(N, 256), 256, 0, stream>>>(dinv, N);

    // --- layer 1: GEMM -> scatter -> (self-loop + bias + stats) -> BN coeffs ---
    gemm_wmma_f32<128, false><<<N / 16, 64, 0, stream>>>(x, W1, nullptr, h, N);
    zero_f32<<<cdiv((long long)NH + 64, 256), 256, 0, stream>>>(agg, NH + 64); // agg + stat sums
    scatter_kernel<<<cdiv((long long)E * 8, 256), 256, 0, stream>>>(rows, cols, dinv, h, agg, E);
    fuse_stats_kernel<<<cdiv((long long)NH, 256), 256, 0, stream>>>(agg, h, dinv, b1, stats, N);
    bn_finalize_kernel<<<1, 32, 0, stream>>>(stats, stats + 64, gamma1, beta1, 1.0f / (float)N);

    // --- layer 2: BN1+ReLU fused into GEMM2's A-loads (agg holds pre-BN values) ---
    gemm_wmma_f32<32, true><<<N / 16, 64, 0, stream>>>(agg, W2, stats + 64, h, N);
    zero_f32<<<cdiv((long long)NH, 256), 256, 0, stream>>>(out, NH);
    zero_f32<<<1, 64, 0, stream>>>(stats, 64);
    scatter_kernel<<<cdiv((long long)E * 8, 256), 256, 0, stream>>>(rows, cols, dinv, h, out, E);
    fuse_stats_kernel<<<cdiv((long long)NH, 256), 256, 0, stream>>>(out, h, dinv, b2, stats, N);
    bn_finalize_kernel<<<1, 32, 0, stream>>>(stats, stats + 64, gamma2, beta2, 1.0f / (float)N);
    bn_apply4_kernel<<<cdiv((long long)(NH / 4), 256), 256, 0, stream>>>(out, stats + 64, NH / 4);
}